// CI3addFrom01_64235530879327
// MI455X (gfx1250) — compile-verified
//
#include <hip/hip_runtime.h>

#define DIM      32
#define BATCH    512
#define M_TOTAL  40704              // 32 + 2*C(32,2) + 8*C(32,3)
#define NCHUNK   64
#define CHUNK    (M_TOTAL / NCHUNK) // 636 exactly

typedef __attribute__((ext_vector_type(2))) float v2f;
typedef __attribute__((ext_vector_type(8))) float v8f;

// ---------------------------------------------------------------------------
// Kernel 1: softmax normalizers. Single block computes max(p) and 1/sum(exp).
// ---------------------------------------------------------------------------
__global__ __launch_bounds__(256) void softmax_scalars_kernel(
    const float* __restrict__ pw, float* __restrict__ sc) {
  __shared__ float red[256];
  const int t = threadIdx.x;

  float lm = -3.402823466e38f;
  for (int k = t; k < M_TOTAL; k += 256) lm = fmaxf(lm, pw[k]);
  red[t] = lm;
  __syncthreads();
  for (int s = 128; s > 0; s >>= 1) {
    if (t < s) red[t] = fmaxf(red[t], red[t + s]);
    __syncthreads();
  }
  const float mx = red[0];
  __syncthreads();

  float ls = 0.0f;
  for (int k = t; k < M_TOTAL; k += 256) ls += __expf(pw[k] - mx);
  red[t] = ls;
  __syncthreads();
  for (int s = 128; s > 0; s >>= 1) {
    if (t < s) red[t] += red[t + s];
    __syncthreads();
  }
  if (t == 0) { sc[0] = mx; sc[1] = 1.0f / red[0]; }
}

// ---------------------------------------------------------------------------
// Kernel 2: main antichain evaluation.
//   threads -> batch elements (512 = 16 wave32s), blocks -> m-chunks.
//   x transposed into LDS: xs[i*512 + b]  (64 KB; gathers are bank-conflict
//   free since lanes differ only in b). idx/preweight reads are wave-uniform
//   -> scalar loads. Each block emits one partial dot per batch element.
// ---------------------------------------------------------------------------
__global__ __launch_bounds__(BATCH) void antichain_kernel(
    const float* __restrict__ x, const float* __restrict__ pw,
    const int* __restrict__ idx, const float* __restrict__ sc,
    float* __restrict__ partials) {
  __shared__ float xs[DIM * BATCH]; // 65536 B, transposed [i][b]
  const int b = threadIdx.x;

  // transpose copy-in: each thread reads its own contiguous row (8x b128
  // loads), scatters to column layout (stride-512 stores hit distinct banks
  // across consecutive lanes).
  const float4* __restrict__ xr4 = (const float4*)(x + b * DIM);
#pragma unroll
  for (int j = 0; j < DIM / 4; ++j) {
    float4 v = xr4[j];
    xs[((4 * j + 0) << 9) + b] = v.x;
    xs[((4 * j + 1) << 9) + b] = v.y;
    xs[((4 * j + 2) << 9) + b] = v.z;
    xs[((4 * j + 3) << 9) + b] = v.w;
  }
  __syncthreads();

  const float mx      = sc[0];
  const float inv_sum = sc[1];
  const int   m0      = blockIdx.x * CHUNK;

  float acc = 0.0f;
  for (int mm = 0; mm < CHUNK; ++mm) {
    const int m = m0 + mm;
    const int* __restrict__ q = idx + m * 9;   // uniform -> s_load
    const float w = __expf(pw[m] - mx) * inv_sum;

    float g0 = fminf(fminf(xs[(q[0] << 9) + b], xs[(q[1] << 9) + b]),
                     xs[(q[2] << 9) + b]);
    float g1 = fminf(fminf(xs[(q[3] << 9) + b], xs[(q[4] << 9) + b]),
                     xs[(q[5] << 9) + b]);
    float g2 = fminf(fminf(xs[(q[6] << 9) + b], xs[(q[7] << 9) + b]),
                     xs[(q[8] << 9) + b]);

    acc = fmaf(w, fmaxf(fmaxf(g0, g1), g2), acc);
  }
  partials[blockIdx.x * BATCH + b] = acc;
}

// ---------------------------------------------------------------------------
// Kernel 3: cross-chunk reduction via WMMA.
//   out[b] = sum_c P[c][b]  ==  ones(16x4) x P-tile(4x16), accumulated over
//   16 k-steps (64 chunk rows). One wave per 16 batch columns.
//   B layout (ISA 7.12.2, 32-bit B 4x16): v0 = rows K=0 (lanes 0-15) and
//   K=2 (lanes 16-31); v1 = rows K=1 / K=3. A = all ones (layout-invariant).
//   D row 0 lives in c[0] of lanes 0-15.
// ---------------------------------------------------------------------------
__global__ __launch_bounds__(32) void reduce_wmma_kernel(
    const float* __restrict__ P, float* __restrict__ out) {
  const int lane = threadIdx.x;        // 0..31
  const int b0   = blockIdx.x * 16;    // 32 blocks x 16 batches = 512
  const int half = lane >> 4;          // 0: rows 0,1   1: rows 2,3
  const int col  = lane & 15;

  v2f a;
  a.x = 1.0f;
  a.y = 1.0f;
  v8f c = {};

#pragma unroll
  for (int k = 0; k < 16; ++k) {
    const int row = 4 * k + 2 * half;
    v2f bv;
    bv.x = P[row * BATCH + b0 + col];
    bv.y = P[(row + 1) * BATCH + b0 + col];
    c = __builtin_amdgcn_wmma_f32_16x16x4_f32(
        /*neg_a=*/false, a, /*neg_b=*/false, bv,
        /*c_mod=*/(short)0, c, /*reuse_a=*/false, /*reuse_b=*/false);
  }

  if (lane < 16) out[b0 + lane] = c[0];
}

// ---------------------------------------------------------------------------
extern "C" void kernel_launch(void* const* d_in, const int* in_sizes, int n_in,
                              void* d_out, int out_size, void* d_ws, size_t ws_size,
                              hipStream_t stream) {
  const float* x   = (const float*)d_in[0];  // (512, 32)  f32
  const float* pw  = (const float*)d_in[1];  // (1, 40704) f32
  const int*   idx = (const int*)d_in[2];    // (40704, 3, 3) i32
  float*       out = (float*)d_out;          // (512, 1)   f32

  float* partials = (float*)d_ws;              // NCHUNK*BATCH floats = 128 KB
  float* sc       = partials + NCHUNK * BATCH; // 2 floats: [max, 1/sum]

  softmax_scalars_kernel<<<1, 256, 0, stream>>>(pw, sc);
  antichain_kernel<<<NCHUNK, BATCH, 0, stream>>>(x, pw, idx, sc, partials);
  reduce_wmma_kernel<<<BATCH / 16, 32, 0, stream>>>(partials, out);
}